// WSMCL_41205916237865
// MI455X (gfx1250) — compile-verified
//
#include <hip/hip_runtime.h>
#include <math.h>

typedef __attribute__((ext_vector_type(2))) float v2f;
typedef __attribute__((ext_vector_type(4))) float v4f;
typedef __attribute__((ext_vector_type(8))) float v8f;

constexpr int Dm = 512;     // model dim
constexpr int Nn = 4096;    // tokens
constexpr int DHd = 64;     // head dim
constexpr int Kk = 10;      // top/bottom K

__device__ __forceinline__ v8f wmma4(v2f a, v2f b, v8f c) {
  // D = A(16x4,f32) * B(4x16,f32) + C(16x16,f32)
  return __builtin_amdgcn_wmma_f32_16x16x4_f32(false, a, false, b, (short)0, c, false, false);
}

__device__ __forceinline__ float gelu_f(float x) {
  return 0.5f * x * (1.0f + erff(x * 0.70710678118654752440f));
}

__device__ __forceinline__ float waveSum(float v) {
#pragma unroll
  for (int o = 16; o > 0; o >>= 1) v += __shfl_xor(v, o, 32);
  return v;
}

// CDNA5 async copies: global -> LDS through the async engine (ASYNCcnt).
// Generic pointers to LDS carry the LDS byte offset in their low 32 bits.
__device__ __forceinline__ void asyncLoadB128(float* ldsDst, const float* g) {
  asm volatile("global_load_async_to_lds_b128 %0, %1, off"
               :: "v"((unsigned)(uintptr_t)ldsDst), "v"(g)
               : "memory");
}
__device__ __forceinline__ void asyncLoadB32(float* ldsDst, const float* g) {
  asm volatile("global_load_async_to_lds_b32 %0, %1, off"
               :: "v"((unsigned)(uintptr_t)ldsDst), "v"(g)
               : "memory");
}
__device__ __forceinline__ void waitAsync0() {
  asm volatile("s_wait_asynccnt 0x0" ::: "memory");
}

// ---------------------------------------------------------------------------
// GEMM: C[4096,512] = op(A[4096,512] @ W[512,512] (+bias) (gelu?)) (+resid)
// block = 256 thr (8 waves); block tile: 16 rows x 512 cols; wave w: cols w*64..
// A row-panel staged once via async b128 (padded stride 514).
// W streamed in 8-row chunks, double-buffered, stored TRANSPOSED in LDS
// (ldsBt[col*12 + krow]) via async b32 scatter -> B fragment = one ds_load_b64.
// storeT=1: write C transposed (C[col][row]) with two float4 stores per tile.
// ---------------------------------------------------------------------------
constexpr int BT_STRIDE = 12;                       // padded cols-major stride
constexpr int GEMM_LDS_FLOATS = 16 * 514 + 2 * Dm * BT_STRIDE;

__global__ void k_gemm(const float* __restrict__ A, const float* __restrict__ W,
                       const float* __restrict__ bias, const float* __restrict__ resid,
                       float* __restrict__ C, int doGelu, int storeT) {
  extern __shared__ float lds[];
  float* ldsA = lds;                       // 16 x 514
  float* ldsBt = lds + 16 * 514;           // 2 x (512 x 12) double buffer
  const int tid = threadIdx.x;
  const int w = tid >> 5, lane = tid & 31;
  const int hh = lane >> 4, r16 = lane & 15;
  const int rowBase = blockIdx.x * 16;
  const int colBase = w * 64;

  {  // stage A row-tile once via async b128 (16B chunks never cross a row)
    const float* Ab = A + (size_t)rowBase * Dm;
#pragma unroll
    for (int it = 0; it < 8; ++it) {
      const int idx = it * 256 + tid;          // 4-float chunk index
      const int r = idx >> 7, c4 = (idx & 127) * 4;
      asyncLoadB128(ldsA + r * 514 + c4, Ab + r * Dm + c4);
    }
  }
  {  // kick off async transposed copy of W chunk 0 into buffer 0
#pragma unroll
    for (int it = 0; it < 16; ++it) {
      const int idx = it * 256 + tid;
      const int krow = idx >> 9, col = idx & 511;
      asyncLoadB32(ldsBt + col * BT_STRIDE + krow, W + idx);
    }
  }
  v8f acc[4];
#pragma unroll
  for (int ct = 0; ct < 4; ++ct) acc[ct] = (v8f){};

  for (int ch = 0; ch < 64; ++ch) {          // K chunks of 8
    waitAsync0();                            // chunk ch (and A) landed in LDS
    __syncthreads();                         // all waves see it; prev buf free
    if (ch + 1 < 64) {                       // async-prefetch next chunk
      const float* Wn = W + (size_t)(ch + 1) * 8 * Dm;
      float* dst = ldsBt + ((ch + 1) & 1) * (Dm * BT_STRIDE);
#pragma unroll
      for (int it = 0; it < 16; ++it) {
        const int idx = it * 256 + tid;
        const int krow = idx >> 9, col = idx & 511;
        asyncLoadB32(dst + col * BT_STRIDE + krow, Wn + idx);
      }
    }
    const float* Bc = ldsBt + (ch & 1) * (Dm * BT_STRIDE);
#pragma unroll
    for (int kb = 0; kb < 2; ++kb) {
      const int k2 = kb * 4 + 2 * hh;
      v2f a = *(const v2f*)(ldsA + r16 * 514 + ch * 8 + k2);
#pragma unroll
      for (int ct = 0; ct < 4; ++ct) {
        const int col = colBase + ct * 16 + r16;
        v2f b = *(const v2f*)(Bc + col * BT_STRIDE + k2);   // contiguous pair
        acc[ct] = wmma4(a, b, acc[ct]);
      }
    }
  }
  if (storeT) {
    // transposed store: C[col][row], 8 consecutive rows per lane per tile
#pragma unroll
    for (int ct = 0; ct < 4; ++ct) {
      const int col = colBase + ct * 16 + r16;
      float* dst = C + (size_t)col * Nn + rowBase + hh * 8;
      v4f lo = {acc[ct][0], acc[ct][1], acc[ct][2], acc[ct][3]};
      v4f hi = {acc[ct][4], acc[ct][5], acc[ct][6], acc[ct][7]};
      *(v4f*)dst = lo;
      *(v4f*)(dst + 4) = hi;
    }
  } else {
#pragma unroll
    for (int ct = 0; ct < 4; ++ct) {
#pragma unroll
      for (int r = 0; r < 8; ++r) {
        const int col = colBase + ct * 16 + r16;
        const int row = rowBase + hh * 8 + r;
        float vv = acc[ct][r];
        if (bias) vv += bias[col];
        if (doGelu) vv = gelu_f(vv);
        if (resid) vv += resid[(size_t)row * Dm + col];
        C[(size_t)row * Dm + col] = vv;
      }
    }
  }
}

// ---------------------------------------------------------------------------
// Row LayerNorm, eps=1e-5, matches jnp two-pass mean/var
// ---------------------------------------------------------------------------
__global__ void k_layernorm(const float* __restrict__ x, const float* __restrict__ g,
                            const float* __restrict__ b, float* __restrict__ out) {
  __shared__ float sh[8];
  const int row = blockIdx.x, t = threadIdx.x;
  const float* xr = x + (size_t)row * Dm;
  const float a0 = xr[t], a1 = xr[t + 256];
  float s = waveSum(a0 + a1);
  if ((t & 31) == 0) sh[t >> 5] = s;
  __syncthreads();
  float mean = 0;
#pragma unroll
  for (int i = 0; i < 8; ++i) mean += sh[i];
  mean *= (1.0f / 512.0f);
  __syncthreads();
  const float d0 = a0 - mean, d1 = a1 - mean;
  float v = waveSum(d0 * d0 + d1 * d1);
  if ((t & 31) == 0) sh[t >> 5] = v;
  __syncthreads();
  float var = 0;
#pragma unroll
  for (int i = 0; i < 8; ++i) var += sh[i];
  var *= (1.0f / 512.0f);
  const float inv = 1.0f / sqrtf(var + 1e-5f);
  out[(size_t)row * Dm + t] = d0 * inv * g[t] + b[t];
  out[(size_t)row * Dm + t + 256] = d1 * inv * g[t + 256] + b[t + 256];
}

__global__ void k_zero(float* __restrict__ p, int n) {
  const int i = blockIdx.x * blockDim.x + threadIdx.x;
  if (i < n) p[i] = 0.0f;
}

// ---------------------------------------------------------------------------
// Fused flash attention (two-pass) + column-sum for attn.mean(axis=(0,1)).
// grid = 8 heads * 32 chunks; block = 8 waves; wave = one 16-row query tile.
// V is consumed pre-transposed: vT[dh_col][n] -> contiguous B fragments.
// ---------------------------------------------------------------------------
__global__ void k_attn(const float* __restrict__ q, const float* __restrict__ kk,
                       const float* __restrict__ vT, float* __restrict__ o,
                       float* __restrict__ colsum) {
  __shared__ float ldsCol[Nn];
  __shared__ float ldsP[8 * 16 * 18];   // per-wave 16x16 P tile, padded stride 18
  const int tid = threadIdx.x;
  const int w = tid >> 5, lane = tid & 31;
  const int hh = lane >> 4, r16 = lane & 15;
  const int head = blockIdx.x >> 5, chunk = blockIdx.x & 31;
  const int h64 = head * DHd;
  const int qbase = chunk * 128 + w * 16;

  for (int i = tid; i < Nn; i += 256) ldsCol[i] = 0.0f;
  __syncthreads();

  // Preload Q fragments (A-layout) for this wave's 16 rows (reused 512 times)
  v2f qa[16];
  {
    const float* qrow = q + (size_t)(qbase + r16) * Dm + h64 + 2 * hh;
#pragma unroll
    for (int i = 0; i < 16; ++i) qa[i] = *(const v2f*)(qrow + i * 4);
  }

  float Mr[8], Lr[8];
#pragma unroll
  for (int r = 0; r < 8; ++r) { Mr[r] = -3.3e38f; Lr[r] = 0.0f; }

  // ------ pass 1: row max & sum of exp ------
  for (int mt = 0; mt < Nn / 16; ++mt) {
    const int mbase = mt * 16;
    const float* krow = kk + (size_t)(mbase + r16) * Dm + h64 + 2 * hh;
    if (mt + 1 < Nn / 16) __builtin_prefetch(krow + 16 * Dm, 0, 3);
    v8f s = (v8f){};
#pragma unroll
    for (int i = 0; i < 16; ++i) {
      v2f b = *(const v2f*)(krow + i * 4);
      s = wmma4(qa[i], b, s);   // S tile: lane holds col=r16, rows 8hh+r
    }
#pragma unroll
    for (int r = 0; r < 8; ++r) {
      const float e = s[r] * 0.125f;   // 1/sqrt(64)
      float tm = e;
#pragma unroll
      for (int off = 1; off < 16; off <<= 1) tm = fmaxf(tm, __shfl_xor(tm, off, 32));
      const float nm = fmaxf(Mr[r], tm);
      float se = expf(e - nm);
#pragma unroll
      for (int off = 1; off < 16; off <<= 1) se += __shfl_xor(se, off, 32);
      Lr[r] = Lr[r] * expf(Mr[r] - nm) + se;
      Mr[r] = nm;
    }
  }
  float invL[8];
#pragma unroll
  for (int r = 0; r < 8; ++r) invL[r] = 1.0f / Lr[r];

  // ------ pass 2: O = P @ V, colsum += sum_rows(P) ------
  v8f oa[4];
#pragma unroll
  for (int ct = 0; ct < 4; ++ct) oa[ct] = (v8f){};
  float* Pw = ldsP + w * (16 * 18);

  // per-lane vT row bases: each lane streams sequentially along n
  const float* vp[4];
#pragma unroll
  for (int ct = 0; ct < 4; ++ct)
    vp[ct] = vT + (size_t)(h64 + ct * 16 + r16) * Nn + 2 * hh;

  for (int mt = 0; mt < Nn / 16; ++mt) {
    const int mbase = mt * 16;
    const float* krow = kk + (size_t)(mbase + r16) * Dm + h64 + 2 * hh;
    if (mt + 1 < Nn / 16) {
      __builtin_prefetch(krow + 16 * Dm, 0, 3);
      __builtin_prefetch(vp[0] + mbase + 16, 0, 3);
    }
    v8f s = (v8f){};
#pragma unroll
    for (int i = 0; i < 16; ++i) {
      v2f b = *(const v2f*)(krow + i * 4);
      s = wmma4(qa[i], b, s);
    }
    float cs = 0.0f;
#pragma unroll
    for (int r = 0; r < 8; ++r) {
      const float e = s[r] * 0.125f;
      const float p = expf(e - Mr[r]) * invL[r];
      Pw[(hh * 8 + r) * 18 + r16] = p;   // store C-layout -> [row][col]
      cs += p;
    }
    cs += __shfl_xor(cs, 16, 32);
    if (hh == 0) atomicAdd(&ldsCol[mbase + r16], cs);

    // Re-read P in A-layout (same-wave LDS ops are in order; no barrier needed)
#pragma unroll
    for (int kb = 0; kb < 4; ++kb) {
      v2f a = *(const v2f*)(Pw + r16 * 18 + kb * 4 + 2 * hh);
#pragma unroll
      for (int ct = 0; ct < 4; ++ct) {
        v2f b = *(const v2f*)(vp[ct] + mbase + kb * 4);   // contiguous pair
        oa[ct] = wmma4(a, b, oa[ct]);
      }
    }
  }
#pragma unroll
  for (int ct = 0; ct < 4; ++ct)
#pragma unroll
    for (int r = 0; r < 8; ++r)
      o[(size_t)(qbase + hh * 8 + r) * Dm + h64 + ct * 16 + r16] = oa[ct][r];

  __syncthreads();
  for (int i = tid; i < Nn; i += 256) atomicAdd(&colsum[i], ldsCol[i]);
}

// ---------------------------------------------------------------------------
// Column mean of feats (4096x512) -> tok[512]
// ---------------------------------------------------------------------------
__global__ void k_colmean(const float* __restrict__ feats, float* __restrict__ tok) {
  const int j = blockIdx.x * 256 + threadIdx.x;
  float s = 0.0f;
  for (int n = 0; n < Nn; ++n) s += feats[(size_t)n * Dm + j];
  tok[j] = s * (1.0f / (float)Nn);
}

// ---------------------------------------------------------------------------
// Top-K / bottom-K (argsort(-A) semantics, stable-tie faithful) + gather atte
// ---------------------------------------------------------------------------
__global__ void k_select(const float* __restrict__ colsum, const float* __restrict__ feats,
                         float* __restrict__ atte) {
  __shared__ float sV[256];
  __shared__ int sI[256];
  __shared__ int picked[2 * Kk];
  __shared__ int pickedBot[Kk];
  const int tid = threadIdx.x;

  // top-K: max, tie -> smaller index
  for (int sel = 0; sel < Kk; ++sel) {
    float bv = -3.4e38f; int bi = 0x7fffffff;
    for (int i = tid; i < Nn; i += 256) {
      bool skip = false;
      for (int p = 0; p < sel; ++p) if (picked[p] == i) skip = true;
      if (skip) continue;
      const float v = colsum[i];
      if (v > bv || (v == bv && i < bi)) { bv = v; bi = i; }
    }
    sV[tid] = bv; sI[tid] = bi;
    __syncthreads();
    for (int off = 128; off > 0; off >>= 1) {
      if (tid < off) {
        const float v2 = sV[tid + off]; const int i2 = sI[tid + off];
        if (v2 > sV[tid] || (v2 == sV[tid] && i2 < sI[tid])) { sV[tid] = v2; sI[tid] = i2; }
      }
      __syncthreads();
    }
    if (tid == 0) picked[sel] = sI[0];
    __syncthreads();
  }
  // bottom-K: sel-th pick = rank N-1-sel (min, tie -> larger index)
  for (int sel = 0; sel < Kk; ++sel) {
    float bv = 3.4e38f; int bi = -1;
    for (int i = tid; i < Nn; i += 256) {
      bool skip = false;
      for (int p = 0; p < sel; ++p) if (pickedBot[p] == i) skip = true;
      if (skip) continue;
      const float v = colsum[i];
      if (v < bv || (v == bv && i > bi)) { bv = v; bi = i; }
    }
    sV[tid] = bv; sI[tid] = bi;
    __syncthreads();
    for (int off = 128; off > 0; off >>= 1) {
      if (tid < off) {
        const float v2 = sV[tid + off]; const int i2 = sI[tid + off];
        if (v2 < sV[tid] || (v2 == sV[tid] && i2 > sI[tid])) { sV[tid] = v2; sI[tid] = i2; }
      }
      __syncthreads();
    }
    if (tid == 0) pickedBot[sel] = sI[0];
    __syncthreads();
  }
  if (tid == 0)
    for (int s = 0; s < Kk; ++s) picked[Kk + (Kk - 1 - s)] = pickedBot[s];
  __syncthreads();
  for (int rr = 0; rr < 2 * Kk; ++rr) {
    const int src = picked[rr];
    for (int j = tid; j < Dm; j += 256)
      atte[rr * Dm + j] = feats[(size_t)src * Dm + j];
  }
}

// ---------------------------------------------------------------------------
// Tail: classifier heads, fused MLP, contrastive loss; writes out[16]
// ---------------------------------------------------------------------------
__global__ void k_tail(const float* __restrict__ tokHE, const float* __restrict__ tokIHC,
                       const float* __restrict__ atteHE, const float* __restrict__ atteIHC,
                       const float* __restrict__ W_he_cls, const float* __restrict__ b_he_cls,
                       const float* __restrict__ W_ihc_cls, const float* __restrict__ b_ihc_cls,
                       const float* __restrict__ W_fc, const float* __restrict__ b_fc,
                       const float* __restrict__ W_c1, const float* __restrict__ b_c1,
                       const float* __restrict__ W_c2, const float* __restrict__ b_c2,
                       const float* __restrict__ logit_scale, float* __restrict__ out) {
  __shared__ float tH[512], tI[512], fused[1536], g1[512];
  __shared__ float Lm[400], rnA[20], rnB[20];
  const int tid = threadIdx.x;
  for (int j = tid; j < 512; j += 256) { tH[j] = tokHE[j]; tI[j] = tokIHC[j]; }
  __syncthreads();
  for (int j = tid; j < 512; j += 256) {
    float s = b_fc[j];
    for (int d = 0; d < 512; ++d) s += tH[d] * W_fc[(size_t)d * 512 + j];
    for (int d = 0; d < 512; ++d) s += tI[d] * W_fc[(size_t)(512 + d) * 512 + j];
    fused[1024 + j] = s;
    fused[j] = tH[j] + tI[j];
    fused[512 + j] = tH[j] * tI[j];
  }
  __syncthreads();
  for (int j = tid; j < 512; j += 256) {
    float s = b_c1[j];
    for (int d = 0; d < 1536; ++d) s += fused[d] * W_c1[(size_t)d * 512 + j];
    g1[j] = gelu_f(s);
  }
  for (int p = tid; p < 40; p += 256) {
    const float* row = (p < 20) ? (atteHE + p * 512) : (atteIHC + (p - 20) * 512);
    float s = 0.0f;
    for (int d = 0; d < 512; ++d) { const float x = row[d]; s += x * x; }
    const float inv = 1.0f / sqrtf(s);
    if (p < 20) rnA[p] = inv; else rnB[p - 20] = inv;
  }
  __syncthreads();
  const float scale = expf(logit_scale[0]);
  for (int p = tid; p < 400; p += 256) {
    const int a = p / 20, b2 = p % 20;
    const float* ra = atteHE + a * 512;
    const float* rb = atteIHC + b2 * 512;
    float s = 0.0f;
    for (int d = 0; d < 512; ++d) s += ra[d] * rb[d];
    Lm[p] = scale * s * rnA[a] * rnB[b2];
  }
  if (tid < 4) {
    float s = b_he_cls[tid];
    for (int d = 0; d < 512; ++d) s += tH[d] * W_he_cls[d * 4 + tid];
    out[tid] = s;
  } else if (tid < 8) {
    const int c = tid - 4;
    float s = b_ihc_cls[c];
    for (int d = 0; d < 512; ++d) s += tI[d] * W_ihc_cls[d * 4 + c];
    out[4 + c] = s;
  }
  __syncthreads();
  if (tid >= 8 && tid < 12) {
    const int c = tid - 8;
    float s = b_c2[c];
    for (int j = 0; j < 512; ++j) s += g1[j] * W_c2[j * 4 + c];
    out[8 + c] = s;
  }
  if (tid == 0) {
    auto M = [&](int r, int c, bool tr) { return tr ? Lm[c * 20 + r] : Lm[r * 20 + c]; };
    float total = 0.0f;
    for (int t = 0; t < 2; ++t) {
      const bool tr = (t == 1);
      float half = 0.0f;
      for (int i = 0; i < Kk; ++i) {
        float s1 = 0.0f;
        for (int r = 0; r < Kk; ++r) {
          const float v0 = M(r, i, tr);
          float mx = v0;
          for (int j = Kk; j < 2 * Kk; ++j) mx = fmaxf(mx, M(r, j, tr));
          float se = expf(v0 - mx);
          for (int j = Kk; j < 2 * Kk; ++j) se += expf(M(r, j, tr) - mx);
          s1 += mx + logf(se) - v0;
        }
        float s2 = 0.0f;
        for (int r = Kk; r < 2 * Kk; ++r) {
          const float v0 = M(r, Kk + i, tr);
          float mx = v0;
          for (int j = 0; j < Kk; ++j) mx = fmaxf(mx, M(r, j, tr));
          float se = expf(v0 - mx);
          for (int j = 0; j < Kk; ++j) se += expf(M(r, j, tr) - mx);
          s2 += mx + logf(se) - v0;
        }
        half += s1 * 0.1f + s2 * 0.1f;
      }
      total += half;
    }
    const float closs = total / (4.0f * (float)Kk);
    out[12] = closs; out[13] = closs; out[14] = closs; out[15] = closs;
  }
}

// ---------------------------------------------------------------------------
extern "C" void kernel_launch(void* const* d_in, const int* in_sizes, int n_in,
                              void* d_out, int out_size, void* d_ws, size_t ws_size,
                              hipStream_t stream) {
  (void)in_sizes; (void)n_in; (void)out_size; (void)ws_size;
  const float* HE_x     = (const float*)d_in[0];
  const float* IHC_x    = (const float*)d_in[1];
  const float* W_he_fc  = (const float*)d_in[2];
  const float* b_he_fc  = (const float*)d_in[3];
  const float* W_ihc_fc = (const float*)d_in[4];
  const float* b_ihc_fc = (const float*)d_in[5];
  const float* he_ln_g  = (const float*)d_in[6];
  const float* he_ln_b  = (const float*)d_in[7];
  const float* he_Wq    = (const float*)d_in[8];
  const float* he_Wk    = (const float*)d_in[9];
  const float* he_Wv    = (const float*)d_in[10];
  const float* he_Wo    = (const float*)d_in[11];
  const float* ihc_ln_g = (const float*)d_in[12];
  const float* ihc_ln_b = (const float*)d_in[13];
  const float* ihc_Wq   = (const float*)d_in[14];
  const float* ihc_Wk   = (const float*)d_in[15];
  const float* ihc_Wv   = (const float*)d_in[16];
  const float* ihc_Wo   = (const float*)d_in[17];
  const float* norm_g   = (const float*)d_in[18];
  const float* norm_b   = (const float*)d_in[19];
  const float* W_he_cls = (const float*)d_in[20];
  const float* b_he_cls = (const float*)d_in[21];
  const float* W_ihc_cls= (const float*)d_in[22];
  const float* b_ihc_cls= (const float*)d_in[23];
  const float* W_fc     = (const float*)d_in[24];
  const float* b_fc     = (const float*)d_in[25];
  const float* W_c1     = (const float*)d_in[26];
  const float* b_c1     = (const float*)d_in[27];
  const float* W_c2     = (const float*)d_in[28];
  const float* b_c2     = (const float*)d_in[29];
  const float* logit_sc = (const float*)d_in[30];

  float* Wsp = (float*)d_ws;
  const size_t ND = (size_t)Nn * Dm;
  float* xg    = Wsp;
  float* h     = Wsp + ND;
  float* q     = Wsp + 2 * ND;
  float* kbuf  = Wsp + 3 * ND;
  float* vbufT = Wsp + 4 * ND;       // V stored transposed: [512][4096]
  float* obuf  = Wsp + 5 * ND;
  float* x2    = h;      // reuse (h dead after q/k/v GEMMs)
  float* feats = q;      // reuse (q dead after attention)
  float* colsum = Wsp + 6 * ND;
  float* tokA   = colsum + Nn;        // 2 * 512
  float* atteA  = tokA + 1024;        // 2 * 20 * 512

  const unsigned gemmLds = GEMM_LDS_FLOATS * sizeof(float);

  for (int b = 0; b < 2; ++b) {
    const float* X   = b ? IHC_x    : HE_x;
    const float* Wfc = b ? W_ihc_fc : W_he_fc;
    const float* bfc = b ? b_ihc_fc : b_he_fc;
    const float* lng = b ? ihc_ln_g : he_ln_g;
    const float* lnb = b ? ihc_ln_b : he_ln_b;
    const float* Wq  = b ? ihc_Wq   : he_Wq;
    const float* Wk  = b ? ihc_Wk   : he_Wk;
    const float* Wv  = b ? ihc_Wv   : he_Wv;
    const float* Wo  = b ? ihc_Wo   : he_Wo;
    float* tok  = tokA + b * 512;
    float* atte = atteA + b * (2 * Kk * Dm);

    k_gemm<<<256, 256, gemmLds, stream>>>(X, Wfc, bfc, nullptr, xg, 1, 0);
    k_layernorm<<<Nn, 256, 0, stream>>>(xg, lng, lnb, h);
    k_gemm<<<256, 256, gemmLds, stream>>>(h, Wq, nullptr, nullptr, q, 0, 0);
    k_gemm<<<256, 256, gemmLds, stream>>>(h, Wk, nullptr, nullptr, kbuf, 0, 0);
    k_gemm<<<256, 256, gemmLds, stream>>>(h, Wv, nullptr, nullptr, vbufT, 0, 1);
    k_zero<<<16, 256, 0, stream>>>(colsum, Nn);
    k_attn<<<256, 256, 0, stream>>>(q, kbuf, vbufT, obuf, colsum);
    k_gemm<<<256, 256, gemmLds, stream>>>(obuf, Wo, nullptr, xg, x2, 0, 0);
    k_layernorm<<<Nn, 256, 0, stream>>>(x2, norm_g, norm_b, feats);
    k_colmean<<<2, 256, 0, stream>>>(feats, tok);
    k_select<<<1, 256, 0, stream>>>(colsum, feats, atte);
  }
  k_tail<<<1, 256, 0, stream>>>(tokA, tokA + 512, atteA, atteA + 2 * Kk * Dm,
                                W_he_cls, b_he_cls, W_ihc_cls, b_ihc_cls,
                                W_fc, b_fc, W_c1, b_c1, W_c2, b_c2,
                                logit_sc, (float*)d_out);
}